// OnlineTripletLoss_7172595384725
// MI455X (gfx1250) — compile-verified
//
#include <hip/hip_runtime.h>
#include <math.h>

#define TL_N 8192
#define TL_D 128
#define TL_MARGIN 0.3f
#define LDS_STRIDE 132          // 128 + 4 floats pad -> conflict-free ds_load_b64
#define CB 1024                 // columns per block
#define TILES (CB / 16)

typedef __attribute__((ext_vector_type(2))) float v2f;
typedef __attribute__((ext_vector_type(8))) float v8f;

// Monotonic float->int mapping (exact total order for non-NaN floats).
__device__ __forceinline__ int map_ord(float f) {
  int i = __float_as_int(f);
  return (i < 0) ? (i ^ 0x7FFFFFFF) : i;
}
__device__ __forceinline__ float unmap_ord(int i) {
  return __int_as_float((i < 0) ? (i ^ 0x7FFFFFFF) : i);
}

// ---------------------------------------------------------------------------
// Kernel 1: per-row squared norms + init atomic merge buffers.
// ---------------------------------------------------------------------------
__global__ void tl_prep(const float* __restrict__ emb, float* __restrict__ sqn,
                        int* __restrict__ hp, unsigned* __restrict__ hn) {
  int row = blockIdx.x * blockDim.x + threadIdx.x;
  if (row >= TL_N) return;
  const float4* p = (const float4*)(emb + (size_t)row * TL_D);
  float s = 0.f;
#pragma unroll
  for (int k = 0; k < TL_D / 4; ++k) {
    float4 v = p[k];
    s += v.x * v.x + v.y * v.y + v.z * v.z + v.w * v.w;
  }
  sqn[row] = s;
  hp[row]  = 0x807FFFFF;   // map_ord(-inf): identity for atomicMax
  hn[row]  = 0x7F800000u;  // +inf bits: identity for uint atomicMin (d >= 0)
}

// ---------------------------------------------------------------------------
// Kernel 2: fused Gram tile (f32 WMMA) -> distances -> masked max/min.
// grid = (TL_N/64, TL_N/CB), block = 128 (4 waves). Wave w owns rows
// [bx*64 + w*16, +16); loops over TILES 16-column tiles of this col-block.
// ---------------------------------------------------------------------------
__global__ __launch_bounds__(128)
void tl_main(const float* __restrict__ emb, const int* __restrict__ labels,
             const float* __restrict__ sqn, int* __restrict__ hp,
             unsigned* __restrict__ hn) {
  __shared__ float ldsB[16 * LDS_STRIDE];

  const int tid  = threadIdx.x;
  const int lane = tid & 31;
  const int wave = tid >> 5;
  const int l16  = lane & 15;
  const int hi   = lane >> 4;

  const int rowBase = blockIdx.x * 64 + wave * 16;
  const int colBase = blockIdx.y * CB;

  // A operand (16x4 f32 WMMA layout): lane holds row (rowBase + l16),
  // k-chunk c covers k = 4c + 2*hi + {0,1}. 32 chunks = full K=128 panel.
  v2f a[32];
  {
    const float* arow = emb + (size_t)(rowBase + l16) * TL_D + 2 * hi;
#pragma unroll
    for (int c = 0; c < 32; ++c)
      a[c] = *(const v2f*)(arow + 4 * c);
  }

  // Per-lane row metadata for the C layout: VGPR r -> row rowBase + r + 8*hi.
  float sn_i[8];
  int   lab_i[8];
#pragma unroll
  for (int r = 0; r < 8; ++r) {
    int i = rowBase + r + 8 * hi;
    sn_i[r]  = sqn[i];
    lab_i[r] = labels[i];
  }

  float mp[8], mn[8];
#pragma unroll
  for (int r = 0; r < 8; ++r) { mp[r] = -INFINITY; mn[r] = INFINITY; }

  for (int t = 0; t < TILES; ++t) {
    const int n0 = colBase + t * 16;

    __syncthreads();
    // Cooperative stage: 16 rows x 128 f32 (8KB) of the column tile into LDS.
#pragma unroll
    for (int k = 0; k < 4; ++k) {
      int fid = tid + 128 * k;    // float4 id 0..511 (32 per row)
      int r   = fid >> 5;
      int c4  = fid & 31;
      float4 v = *(const float4*)(emb + (size_t)(n0 + r) * TL_D + c4 * 4);
      *(float4*)(&ldsB[r * LDS_STRIDE + c4 * 4]) = v;
    }
    if (t + 1 < TILES) {  // pull next tile toward L2/L0 (global_prefetch_b8)
      const float* nxt =
          emb + (size_t)(colBase + (t + 1) * 16 + (tid >> 3)) * TL_D + (tid & 7) * 16;
      __builtin_prefetch(nxt, 0, 0);
    }
    __syncthreads();

    // Gram tile: C(16x16) = A(16x128) * B(128x16), 32 x v_wmma_f32_16x16x4_f32.
    // B layout mirrors A (B[k][n] = X[n][k]): lane reads LDS row l16,
    // k = 4c + 2*hi + {0,1}.
    v8f acc = {};
    const float* brow = &ldsB[l16 * LDS_STRIDE + 2 * hi];
#pragma unroll
    for (int c = 0; c < 32; ++c) {
      v2f b = *(const v2f*)(brow + 4 * c);
      acc = __builtin_amdgcn_wmma_f32_16x16x4_f32(
          false, a[c], false, b, (short)0, acc, false, false);
    }

    // Epilogue: distance + masks folded into running max/min.
    const int   j     = n0 + l16;
    const float sn_j  = sqn[j];
    const int   lab_j = labels[j];
#pragma unroll
    for (int r = 0; r < 8; ++r) {
      const int i = rowBase + r + 8 * hi;
      float sq = sn_i[r] + sn_j - 2.0f * acc[r];
      sq = fmaxf(sq, 0.0f);
      float d = (sq > 0.0f) ? sqrtf(sq) : 0.0f;
      bool same = (lab_i[r] == lab_j);
      bool diag = (i == j);
      mp[r] = fmaxf(mp[r], (same && !diag) ? d : -INFINITY);
      mn[r] = fminf(mn[r], (!same) ? d : INFINITY);
    }
  }

  // Butterfly reduce across the 16 lanes sharing each row, then merge
  // across column-blocks with exact ordered atomics.
#pragma unroll
  for (int r = 0; r < 8; ++r) {
    float p = mp[r], q = mn[r];
#pragma unroll
    for (int m = 1; m < 16; m <<= 1) {
      p = fmaxf(p, __shfl_xor(p, m, 16));
      q = fminf(q, __shfl_xor(q, m, 16));
    }
    if (l16 == 0) {
      const int i = rowBase + r + 8 * hi;
      atomicMax(&hp[i], map_ord(p));
      atomicMin(&hn[i], __float_as_uint(q));
    }
  }
}

// ---------------------------------------------------------------------------
// Kernel 3: masked mean over rows -> scalar loss.
// ---------------------------------------------------------------------------
__global__ void tl_final(const int* __restrict__ hp,
                         const unsigned* __restrict__ hn,
                         float* __restrict__ out) {
  __shared__ float ssum[256];
  __shared__ int   scnt[256];
  int tid = threadIdx.x;
  float sum = 0.f;
  int   cnt = 0;
  for (int i = tid; i < TL_N; i += 256) {
    float hpf = unmap_ord(hp[i]);
    float hnf = __uint_as_float(hn[i]);
    bool valid = (hpf > -INFINITY) && (hnf < INFINITY);
    float per = fmaxf(hpf - hnf + TL_MARGIN, 0.f);
    sum += valid ? per : 0.f;
    cnt += valid ? 1 : 0;
  }
  ssum[tid] = sum;
  scnt[tid] = cnt;
  __syncthreads();
  for (int s = 128; s > 0; s >>= 1) {
    if (tid < s) { ssum[tid] += ssum[tid + s]; scnt[tid] += scnt[tid + s]; }
    __syncthreads();
  }
  if (tid == 0) out[0] = (scnt[0] > 0) ? (ssum[0] / (float)scnt[0]) : 0.f;
}

// ---------------------------------------------------------------------------
extern "C" void kernel_launch(void* const* d_in, const int* in_sizes, int n_in,
                              void* d_out, int out_size, void* d_ws, size_t ws_size,
                              hipStream_t stream) {
  const float* emb    = (const float*)d_in[0];
  const int*   labels = (const int*)d_in[1];

  float*    sqn = (float*)d_ws;
  int*      hp  = (int*)((char*)d_ws + (size_t)TL_N * sizeof(float));
  unsigned* hn  = (unsigned*)((char*)d_ws + 2ull * TL_N * sizeof(float));
  float*    out = (float*)d_out;

  tl_prep<<<TL_N / 256, 256, 0, stream>>>(emb, sqn, hp, hn);
  dim3 grid(TL_N / 64, TL_N / CB);
  tl_main<<<grid, 128, 0, stream>>>(emb, labels, sqn, hp, hn);
  tl_final<<<1, 256, 0, stream>>>(hp, hn, out);
}